// MultiHeadHashRetrieval_30846455120558
// MI455X (gfx1250) — compile-verified
//
#include <hip/hip_runtime.h>

// ---------------------------------------------------------------------------
// MultiHeadHashRetrieval for MI455X (gfx1250)
//
// out[b,s, t*64:(t+1)*64] = W[t, idx_t(b,s)],  t = (n-2)*4 + k, n in {2,3,4}
// idx = ((sum_j g_j * base_k^j) % modulus_k) % 500000   (exact int64)
//
// Memory-bound gather: ~203 MB traffic -> ~9 us floor at 23.3 TB/s.
// CDNA5 path: ASYNCcnt-tracked global->LDS gather + LDS->global store,
// 2 positions per wave (12 outstanding 256B-row gathers), branch-free
// Barrett-reduction hashing to keep VALU/SALU off the critical path.
// ---------------------------------------------------------------------------

// ---- compile-time reproduction of the reference's _prime() constants ------
constexpr bool cx_is_prime(long long x) {
  if (x < 2) return false;
  for (long long i = 2; i * i <= x; ++i)
    if (x % i == 0) return false;
  return true;
}
constexpr long long cx_next_prime(long long n) {
  while (!cx_is_prime(n)) ++n;
  return n;
}

constexpr unsigned BASEV[4] = {
    (unsigned)cx_next_prime(31),   // 31
    (unsigned)cx_next_prime(131),  // 131
    (unsigned)cx_next_prime(231),  // 233
    (unsigned)cx_next_prime(331)}; // 331
constexpr unsigned MODV[4] = {
    (unsigned)cx_next_prime(500000),
    (unsigned)cx_next_prime(501000),
    (unsigned)cx_next_prime(502000),
    (unsigned)cx_next_prime(503000)};
// floor(2^64 / M) == floor((2^64-1)/M) for odd prime M.
// For h < 2^41: q = umulhi64(h, MAGIC) is floor(h/M) or floor(h/M)-1,
// fixed by one conditional subtract.
constexpr unsigned long long MAGICV[4] = {
    ~0ULL / MODV[0], ~0ULL / MODV[1], ~0ULL / MODV[2], ~0ULL / MODV[3]};

constexpr int TABLE    = 500000;
constexpr int DIM      = 64;
constexpr int N_TABLES = 12;
constexpr int OUT_DIM  = N_TABLES * DIM;   // 768 floats per position
constexpr int WAVES_PER_BLOCK = 8;         // 256 threads, wave32
constexpr int POS_PER_WAVE    = 2;

// Branch-free per-lane hash for table t (lane-private), position p.
static __device__ __forceinline__ int hash_row(
    const int* __restrict__ ng2, const int* __restrict__ ng3,
    const int* __restrict__ ng4, int p, int t)
{
  const int k    = t & 3;          // hash head
  const int nlen = 2 + (t >> 2);   // n-gram length 2..4

  // pointer select via cndmask (no branches)
  const int* gp = ng2 + (size_t)p * 2;
  gp = (nlen == 3) ? (ng3 + (size_t)p * 3) : gp;
  gp = (nlen == 4) ? (ng4 + (size_t)p * 4) : gp;

  // padded 4-term Horner: clamped in-bounds loads, zero-masked high terms
  const unsigned long long g0 = (unsigned)gp[0];
  const unsigned long long g1 = (unsigned)gp[1];
  unsigned long long g2 = (unsigned)gp[(nlen >= 3) ? 2 : 1];
  unsigned long long g3 = (unsigned)gp[nlen - 1];
  g2 = (nlen >= 3) ? g2 : 0ULL;
  g3 = (nlen == 4) ? g3 : 0ULL;

  const unsigned long long b =
      (k == 0) ? BASEV[0] : (k == 1) ? BASEV[1] : (k == 2) ? BASEV[2] : BASEV[3];
  const unsigned m =
      (k == 0) ? MODV[0] : (k == 1) ? MODV[1] : (k == 2) ? MODV[2] : MODV[3];
  const unsigned long long magic =
      (k == 0) ? MAGICV[0] : (k == 1) ? MAGICV[1] : (k == 2) ? MAGICV[2] : MAGICV[3];

  const unsigned long long h = ((g3 * b + g2) * b + g1) * b + g0; // < 2^41

  // Barrett: h % m with one correction
  unsigned long long q = __umul64hi(h, magic);
  unsigned r = (unsigned)(h - q * m);
  r = (r >= m) ? r - m : r;
  // r < m < 2*TABLE  ->  % TABLE is one compare-subtract
  r = (r >= (unsigned)TABLE) ? r - (unsigned)TABLE : r;
  return (int)r;
}

__global__ __launch_bounds__(256)
void mhh_gather_kernel(const int* __restrict__ ng2,
                       const int* __restrict__ ng3,
                       const int* __restrict__ ng4,
                       const float* __restrict__ W,
                       float* __restrict__ out,
                       int P)
{
  // 8 waves * 2 positions * 768 floats = 48 KB LDS staging (WGP has 320 KB)
  __shared__ __align__(16) float stage[WAVES_PER_BLOCK * POS_PER_WAVE * OUT_DIM];

  const int lane  = threadIdx.x & 31;
  const int wave  = threadIdx.x >> 5;
  const int pbase = (blockIdx.x * WAVES_PER_BLOCK + wave) * POS_PER_WAVE;
  if (pbase >= P) return;

  // lanes 0..11 each own table t = lane (others duplicate t=0, discarded)
  const int t   = (lane < N_TABLES) ? lane : 0;
  const int sub = lane >> 4;   // which of the 2 rows this B128 op covers
  const int col = lane & 15;   // float4 column within the 64-float row

  const unsigned lds0 =
      (unsigned)(size_t)(const void*)&stage[wave * POS_PER_WAVE * OUT_DIM];

  int rows[POS_PER_WAVE];
  #pragma unroll
  for (int i = 0; i < POS_PER_WAVE; ++i) {
    const int p = pbase + i;
    rows[i] = (p < P) ? hash_row(ng2, ng3, ng4, p, t) : 0;
  }

  // ---- issue all async gathers first: 12 outstanding 512B transfers -------
  #pragma unroll
  for (int i = 0; i < POS_PER_WAVE; ++i) {
    if (pbase + i >= P) break;
    #pragma unroll
    for (int j = 0; j < 6; ++j) {
      const int tt = j * 2 + sub;
      const int r  = __shfl(rows[i], tt, 32);
      const unsigned long long ga = (unsigned long long)(const void*)
          (W + ((size_t)tt * TABLE + (size_t)r) * DIM + (size_t)col * 4);
      const unsigned la = lds0 + (unsigned)((i * 192 + j * 32 + lane) * 16);
      asm volatile("global_load_async_to_lds_b128 %0, %1, off"
                   :: "v"(la), "v"(ga) : "memory");
    }
  }

#if defined(__has_builtin)
#if __has_builtin(__builtin_amdgcn_s_wait_asynccnt)
  __builtin_amdgcn_s_wait_asynccnt(0);
#else
  asm volatile("s_wait_asynccnt 0x0" ::: "memory");
#endif
#else
  asm volatile("s_wait_asynccnt 0x0" ::: "memory");
#endif

  // ---- coalesced async stores: LDS -> global, data never touches VGPRs ----
  // S_ENDPGM's implicit wait-idle drains ASYNCcnt before the wave exits.
  #pragma unroll
  for (int i = 0; i < POS_PER_WAVE; ++i) {
    const int p = pbase + i;
    if (p >= P) break;
    const size_t outbase = (size_t)p * OUT_DIM;
    #pragma unroll
    for (int j = 0; j < 6; ++j) {
      const unsigned long long ga = (unsigned long long)(void*)
          (out + outbase + (size_t)(j * 32 + lane) * 4);
      const unsigned la = lds0 + (unsigned)((i * 192 + j * 32 + lane) * 16);
      asm volatile("global_store_async_from_lds_b128 %0, %1, off"
                   :: "v"(ga), "v"(la) : "memory");
    }
  }
}

extern "C" void kernel_launch(void* const* d_in, const int* in_sizes, int n_in,
                              void* d_out, int out_size, void* d_ws, size_t ws_size,
                              hipStream_t stream) {
  (void)n_in; (void)out_size; (void)d_ws; (void)ws_size;
  const int*   ng2 = (const int*)d_in[0];   // (B,S,2) int
  const int*   ng3 = (const int*)d_in[1];   // (B,S,3) int
  const int*   ng4 = (const int*)d_in[2];   // (B,S,4) int
  const float* W   = (const float*)d_in[3]; // (12, 500000, 64) f32
  float*       out = (float*)d_out;         // (B,S,768) f32

  const int P = in_sizes[0] / 2;            // B*S positions
  const int pos_per_block = WAVES_PER_BLOCK * POS_PER_WAVE;
  const int blocks = (P + pos_per_block - 1) / pos_per_block;

  hipLaunchKernelGGL(mhh_gather_kernel, dim3(blocks), dim3(256), 0, stream,
                     ng2, ng3, ng4, W, out, P);
}